// DETM_53326313947567
// MI455X (gfx1250) — compile-verified
//
#include <hip/hip_runtime.h>
#include <math.h>

// ---------------- problem constants ----------------
#define Vv   10000
#define Tt   100
#define Kk   50
#define Rr   300
#define Hh   200
#define ENe  800
#define Lc   3
#define Bb   128
#define CTc  5                       // time chunk for beta/logit processing
#define NVT  ((Vv + 255) / 256)      // 40 v-tiles for NLL partials
#define KLA_BLOCKS ((Kk * Tt * Rr + 255) / 256)
#define LOG_DELTA (-5.298317367f)    // log(0.005)

typedef __attribute__((ext_vector_type(2))) float v2f;
typedef __attribute__((ext_vector_type(8))) float v8f;

#if defined(__has_builtin)
#  if __has_builtin(__builtin_amdgcn_sched_group_barrier)
#    define SCHED_VMEM_READ(n) __builtin_amdgcn_sched_group_barrier(0x0020, (n), 0)
#    define SCHED_MFMA(n)      __builtin_amdgcn_sched_group_barrier(0x0008, (n), 0)
#  endif
#endif
#ifndef SCHED_VMEM_READ
#  define SCHED_VMEM_READ(n)
#  define SCHED_MFMA(n)
#endif

__device__ __forceinline__ float sigf(float x) { return 1.0f / (1.0f + expf(-x)); }

__device__ __forceinline__ v8f wmma4(v2f a, v2f b, v8f c) {
    return __builtin_amdgcn_wmma_f32_16x16x4_f32(false, a, false, b, (short)0, c, false, false);
}

// ---------------------------------------------------------------
// f32 WMMA GEMM:  C[M,N] = A[M,KD] * W[N,KD]^T (+bias) (+relu)
// Each wave computes a 16x64 C slab (4 accumulators, shared A fragment):
// 5 b64 loads per 4 v_wmma_f32_16x16x4_f32. KD is a template parameter.
// The K-loop is software-pipelined (next step's operands loaded before
// this step's wmmas) and the schedule is pinned with sched_group_barrier
// (5 x VMEM-read then 4 x WMMA per step) so the backend cannot sink the
// loads back to their uses and re-serialize with s_wait_loadcnt 0x0.
// A-frag: lane(0..15)=row M, half=lane>>4 selects K pair {0,1}/{2,3}.
// B-frag: lane=col N, same K-pair split; both are contiguous float2.
// D: VGPR j -> row (j + 8*half), lane -> col.
// ---------------------------------------------------------------
template<int KD>
__global__ void gemm_wmma(const float* __restrict__ A, const float* __restrict__ W,
                          const float* __restrict__ bias, float* __restrict__ C,
                          int M, int N, int relu)
{
    const int lane = threadIdx.x;                        // 0..31
    const int grp  = blockIdx.x * 4 + threadIdx.y;       // 64-col group per wave
    const int col0 = grp * 64;
    const int row0 = blockIdx.y * 16;
    if (col0 >= N) return;                               // wave-uniform exit

    const int half = lane >> 4;
    const int r    = lane & 15;
    int arow = row0 + r; if (arow >= M) arow = M - 1;    // clamp (store guarded)
    int w0 = col0 + r;      if (w0 >= N) w0 = N - 1;
    int w1 = col0 + 16 + r; if (w1 >= N) w1 = N - 1;
    int w2 = col0 + 32 + r; if (w2 >= N) w2 = N - 1;
    int w3 = col0 + 48 + r; if (w3 >= N) w3 = N - 1;

    const float* __restrict__ Ap  = A + (size_t)arow * KD + 2 * half;
    const float* __restrict__ Wp0 = W + (size_t)w0 * KD + 2 * half;
    const float* __restrict__ Wp1 = W + (size_t)w1 * KD + 2 * half;
    const float* __restrict__ Wp2 = W + (size_t)w2 * KD + 2 * half;
    const float* __restrict__ Wp3 = W + (size_t)w3 * KD + 2 * half;

    v8f acc0 = {0.f,0.f,0.f,0.f,0.f,0.f,0.f,0.f};
    v8f acc1 = acc0, acc2 = acc0, acc3 = acc0;

    constexpr int K4 = KD >> 2;                          // full 4-K steps
    // prologue: operands for step 0
    v2f a  = *(const v2f*)(Ap);
    v2f b0 = *(const v2f*)(Wp0);
    v2f b1 = *(const v2f*)(Wp1);
    v2f b2 = *(const v2f*)(Wp2);
    v2f b3 = *(const v2f*)(Wp3);
    for (int i = 0; i < K4 - 1; ++i) {
        // issue next step's loads BEFORE this step's wmmas (double-buffer)
        const v2f an  = *(const v2f*)(Ap  + 4 * (i + 1));
        const v2f bn0 = *(const v2f*)(Wp0 + 4 * (i + 1));
        const v2f bn1 = *(const v2f*)(Wp1 + 4 * (i + 1));
        const v2f bn2 = *(const v2f*)(Wp2 + 4 * (i + 1));
        const v2f bn3 = *(const v2f*)(Wp3 + 4 * (i + 1));
        acc0 = wmma4(a, b0, acc0);
        acc1 = wmma4(a, b1, acc1);
        acc2 = wmma4(a, b2, acc2);
        acc3 = wmma4(a, b3, acc3);
        a = an; b0 = bn0; b1 = bn1; b2 = bn2; b3 = bn3;
        // pin per-iteration schedule: 5 loads first, then 4 wmmas
        SCHED_VMEM_READ(5);
        SCHED_MFMA(4);
    }
    acc0 = wmma4(a, b0, acc0);                           // epilogue step
    acc1 = wmma4(a, b1, acc1);
    acc2 = wmma4(a, b2, acc2);
    acc3 = wmma4(a, b3, acc3);

    if constexpr ((KD & 3) != 0) {                       // K tail (only KD=10050)
        const int kb = (K4 << 2) + 2 * half;
        v2f ta = {0.f,0.f}, tb0 = {0.f,0.f}, tb1 = {0.f,0.f}, tb2 = {0.f,0.f}, tb3 = {0.f,0.f};
        if (kb < KD) {
            ta.x  = Ap[4 * K4];
            tb0.x = Wp0[4 * K4]; tb1.x = Wp1[4 * K4];
            tb2.x = Wp2[4 * K4]; tb3.x = Wp3[4 * K4];
        }
        if (kb + 1 < KD) {
            ta.y  = Ap[4 * K4 + 1];
            tb0.y = Wp0[4 * K4 + 1]; tb1.y = Wp1[4 * K4 + 1];
            tb2.y = Wp2[4 * K4 + 1]; tb3.y = Wp3[4 * K4 + 1];
        }
        acc0 = wmma4(ta, tb0, acc0);
        acc1 = wmma4(ta, tb1, acc1);
        acc2 = wmma4(ta, tb2, acc2);
        acc3 = wmma4(ta, tb3, acc3);
    }

    // store 4 sub-tiles
    v8f accs[4] = {acc0, acc1, acc2, acc3};
#pragma unroll
    for (int s = 0; s < 4; ++s) {
        const int col = col0 + 16 * s + r;
        if (col < N) {
            const float bv = bias ? bias[col] : 0.0f;
#pragma unroll
            for (int j = 0; j < 8; ++j) {
                const int orow = row0 + j + 8 * half;
                if (orow < M) {
                    float v = accs[s][j] + bv;
                    if (relu) v = fmaxf(v, 0.0f);
                    C[(size_t)orow * N + col] = v;
                }
            }
        }
    }
}

// ---------------------------------------------------------------
// 3-layer LSTM over T=100 steps (single block, persistent state in LDS)
// gate order i,f,g,o. Ping-pongs buf0<->buf1; final output lands in buf1.
// ---------------------------------------------------------------
__global__ void lstm_kernel(const float* __restrict__ Wih, const float* __restrict__ Whh,
                            const float* __restrict__ bih, const float* __restrict__ bhh,
                            float* __restrict__ buf0, float* __restrict__ buf1)
{
    __shared__ float xt[Hh], hs[Hh], cs[Hh], gs[4 * Hh];
    const int tid = threadIdx.x;                         // 800 threads
    for (int l = 0; l < Lc; ++l) {
        const float* Wi = Wih + (size_t)l * 4 * Hh * Hh;
        const float* Wh = Whh + (size_t)l * 4 * Hh * Hh;
        const float* bi = bih + l * 4 * Hh;
        const float* bh = bhh + l * 4 * Hh;
        float* inb  = (l & 1) ? buf1 : buf0;
        float* outb = (l & 1) ? buf0 : buf1;
        if (tid < Hh) { hs[tid] = 0.f; cs[tid] = 0.f; }
        __syncthreads();
        for (int t = 0; t < Tt; ++t) {
            if (tid < Hh) xt[tid] = inb[t * Hh + tid];
            __syncthreads();
            {
                const float* wi = Wi + (size_t)tid * Hh;
                const float* wh = Wh + (size_t)tid * Hh;
                float s = bi[tid] + bh[tid];
#pragma unroll 4
                for (int i = 0; i < Hh; ++i) s += wi[i] * xt[i] + wh[i] * hs[i];
                gs[tid] = s;
            }
            __syncthreads();
            if (tid < Hh) {
                const float ig = sigf(gs[tid]);
                const float fg = sigf(gs[Hh + tid]);
                const float gg = tanhf(gs[2 * Hh + tid]);
                const float og = sigf(gs[3 * Hh + tid]);
                const float c  = fg * cs[tid] + ig * gg;
                const float h  = og * tanhf(c);
                cs[tid] = c; hs[tid] = h;
                outb[t * Hh + tid] = h;
            }
            __syncthreads();
        }
    }
}

// ---------------------------------------------------------------
// eta scan: per t, mu/ls = W @ [lstm_out_t, eta_prev] + b ; KL ; eta = mu
// ---------------------------------------------------------------
__global__ void eta_kernel(const float* __restrict__ lstm_out,
                           const float* __restrict__ muW, const float* __restrict__ mub,
                           const float* __restrict__ lsW, const float* __restrict__ lsb,
                           float* __restrict__ etas, float* __restrict__ kl_out)
{
    __shared__ float xb[Hh + Kk], mu[Kk], ls[Kk], eprev[Kk], terms[Kk];
    const int tid = threadIdx.x;                         // 128 threads
    if (tid < Kk) eprev[tid] = 0.f;
    __syncthreads();
    float klacc = 0.f;                                   // thread 0 only
    for (int t = 0; t < Tt; ++t) {
        for (int i = tid; i < Hh; i += 128) xb[i] = lstm_out[t * Hh + i];
        if (tid < Kk) xb[Hh + tid] = eprev[tid];
        __syncthreads();
        if (tid < Kk) {
            const float* w = muW + tid * (Hh + Kk);
            float s = mub[tid];
            for (int i = 0; i < Hh + Kk; ++i) s += w[i] * xb[i];
            mu[tid] = s;
        } else if (tid >= 64 && tid < 64 + Kk) {
            const int k = tid - 64;
            const float* w = lsW + k * (Hh + Kk);
            float s = lsb[k];
            for (int i = 0; i < Hh + Kk; ++i) s += w[i] * xb[i];
            ls[k] = s;
        }
        __syncthreads();
        if (tid < Kk) {
            const float pls = (t == 0) ? 0.f : LOG_DELTA;
            const float pmu = xb[Hh + tid];
            const float qm = mu[tid], ql = ls[tid];
            terms[tid] = 0.5f * ((expf(ql) + (qm - pmu) * (qm - pmu)) /
                                 (expf(pls) + 1e-6f) - 1.f + pls - ql);
            etas[t * Kk + tid] = qm;
            eprev[tid] = qm;
        }
        __syncthreads();
        if (tid == 0) { float s = 0.f; for (int k = 0; k < Kk; ++k) s += terms[k]; klacc += s; }
        __syncthreads();
    }
    if (tid == 0) kl_out[0] = klacc;
}

// ---------------------------------------------------------------
// Xcat[b] = [ bows[b]/sum(bows[b]) , etas[times[b]] ]   (one block per b)
// ---------------------------------------------------------------
__global__ void xcat_kernel(const float* __restrict__ bows, const int* __restrict__ times,
                            const float* __restrict__ etas, float* __restrict__ Xcat)
{
    const int b = blockIdx.x, tid = threadIdx.x;
    __shared__ float red[256];
    __shared__ float inv;
    float s = 0.f;
    for (int v = tid; v < Vv; v += 256) s += bows[(size_t)b * Vv + v];
    red[tid] = s; __syncthreads();
    for (int o = 128; o > 0; o >>= 1) { if (tid < o) red[tid] += red[tid + o]; __syncthreads(); }
    if (tid == 0) inv = 1.f / red[0];
    __syncthreads();
    float* xr = Xcat + (size_t)b * (Vv + Kk);
    for (int v = tid; v < Vv; v += 256) xr[v] = bows[(size_t)b * Vv + v] * inv;
    if (tid < Kk) xr[Vv + tid] = etas[times[b] * Kk + tid];
}

// ---------------------------------------------------------------
// theta = softmax(mu_t), kl_theta per-b partial (one block per b)
// ---------------------------------------------------------------
__global__ void theta_kernel(const float* __restrict__ mu_t, const float* __restrict__ ls_t,
                             const int* __restrict__ times, const float* __restrict__ etas,
                             float* __restrict__ theta, float* __restrict__ klt_part)
{
    const int b = blockIdx.x, tid = threadIdx.x;         // 64 threads
    __shared__ float m[Kk], terms[Kk];
    __shared__ float mx, sm;
    if (tid < Kk) m[tid] = mu_t[b * Kk + tid];
    __syncthreads();
    if (tid == 0) {
        float mm = m[0];
        for (int k = 1; k < Kk; ++k) mm = fmaxf(mm, m[k]);
        float s = 0.f;
        for (int k = 0; k < Kk; ++k) s += expf(m[k] - mm);
        mx = mm; sm = s;
    }
    __syncthreads();
    if (tid < Kk) {
        theta[b * Kk + tid] = expf(m[tid] - mx) / sm;
        const float qm = m[tid], ql = ls_t[b * Kk + tid];
        const float pm = etas[times[b] * Kk + tid];
        terms[tid] = 0.5f * ((expf(ql) + (qm - pm) * (qm - pm)) / (1.f + 1e-6f) - 1.f - ql);
    }
    __syncthreads();
    if (tid == 0) { float s = 0.f; for (int k = 0; k < Kk; ++k) s += terms[k]; klt_part[b] = s; }
}

// ---------------------------------------------------------------
// kl_alpha elementwise over [K,T,R] -> per-block partial
// ---------------------------------------------------------------
__global__ void klalpha_kernel(const float* __restrict__ qmu, const float* __restrict__ qls,
                               float* __restrict__ part)
{
    __shared__ float red[256];
    const int tid = threadIdx.x;
    const size_t idx = (size_t)blockIdx.x * 256 + tid;
    const size_t TOT = (size_t)Kk * Tt * Rr;
    float term = 0.f;
    if (idx < TOT) {
        const size_t rem = idx % ((size_t)Tt * Rr);
        const int t = (int)(rem / Rr);
        const float qm = qmu[idx], ql = qls[idx];
        const float pm  = (t == 0) ? 0.f : qmu[idx - Rr];
        const float pls = (t == 0) ? 0.f : LOG_DELTA;
        term = 0.5f * ((expf(ql) + (qm - pm) * (qm - pm)) /
                       (expf(pls) + 1e-6f) - 1.f + pls - ql);
    }
    red[tid] = term; __syncthreads();
    for (int o = 128; o > 0; o >>= 1) { if (tid < o) red[tid] += red[tid + o]; __syncthreads(); }
    if (tid == 0) part[blockIdx.x] = red[0];
}

// gather alpha rows for a time chunk: Achunk[tl*K+k, r] = mu_q_alpha[k, t0+tl, r]
__global__ void gather_alpha(const float* __restrict__ qmu, float* __restrict__ Achunk, int t0)
{
    const int idx = blockIdx.x * 256 + threadIdx.x;
    const int tot = CTc * Kk * Rr;
    if (idx >= tot) return;
    const int row = idx / Rr, r = idx % Rr;
    const int k = row % Kk, tl = row / Kk;
    Achunk[idx] = qmu[((size_t)k * Tt + t0 + tl) * Rr + r];
}

// per-row softmax stats over V (one block per row of the chunk)
__global__ void rowstats_kernel(const float* __restrict__ logit,
                                float* __restrict__ rmax, float* __restrict__ rinv)
{
    const int row = blockIdx.x, tid = threadIdx.x;       // 256 threads
    __shared__ float red[256];
    __shared__ float mxs;
    const float* lr = logit + (size_t)row * Vv;
    float m = -3.4e38f;
    for (int v = tid; v < Vv; v += 256) m = fmaxf(m, lr[v]);
    red[tid] = m; __syncthreads();
    for (int o = 128; o > 0; o >>= 1) { if (tid < o) red[tid] = fmaxf(red[tid], red[tid + o]); __syncthreads(); }
    if (tid == 0) mxs = red[0];
    __syncthreads();
    const float mx = mxs;
    float s = 0.f;
    for (int v = tid; v < Vv; v += 256) s += expf(lr[v] - mx);
    red[tid] = s; __syncthreads();
    for (int o = 128; o > 0; o >>= 1) { if (tid < o) red[tid] += red[tid + o]; __syncthreads(); }
    if (tid == 0) { rmax[row] = mx; rinv[row] = 1.f / red[0]; }
}

// fused mixture + NLL partial per (b, v-tile); only b's in this time chunk run
__global__ void nll_kernel(const float* __restrict__ logit, const float* __restrict__ rmax,
                           const float* __restrict__ rinv, const float* __restrict__ theta,
                           const float* __restrict__ bows, const int* __restrict__ times,
                           float* __restrict__ nll_part, int t0)
{
    const int b = blockIdx.y;
    const int tb = times[b];
    if (tb < t0 || tb >= t0 + CTc) return;
    const int tl = tb - t0;
    __shared__ float th[Kk], mxl[Kk], ivl[Kk], red[256];
    const int tid = threadIdx.x;
    if (tid < Kk) {
        th[tid]  = theta[b * Kk + tid];
        mxl[tid] = rmax[tl * Kk + tid];
        ivl[tid] = rinv[tl * Kk + tid];
    }
    __syncthreads();
    const int v = blockIdx.x * 256 + tid;
    float val = 0.f;
    if (v < Vv) {
        float s = 0.f;
        const float* lb = logit + (size_t)(tl * Kk) * Vv + v;
#pragma unroll 5
        for (int k = 0; k < Kk; ++k)
            s += th[k] * expf(lb[(size_t)k * Vv] - mxl[k]) * ivl[k];
        val = -logf(s + 1e-12f) * bows[(size_t)b * Vv + v];
    }
    red[tid] = val; __syncthreads();
    for (int o = 128; o > 0; o >>= 1) { if (tid < o) red[tid] += red[tid + o]; __syncthreads(); }
    if (tid == 0) nll_part[b * NVT + blockIdx.x] = red[0];
}

__global__ void zero_kernel(float* p, int n)
{
    const int i = blockIdx.x * 256 + threadIdx.x;
    if (i < n) p[i] = 0.f;
}

// deterministic final reduction -> d_out = {loss, nll, kl_eta, kl_theta, kl_alpha}
__global__ void finalize_kernel(const float* __restrict__ nll_part, const float* __restrict__ klt_part,
                                const float* __restrict__ kla_part, const float* __restrict__ kle,
                                float* __restrict__ out)
{
    __shared__ float red[256];
    const int tid = threadIdx.x;
    float a = 0.f, b = 0.f, c = 0.f;
    for (int i = tid; i < Bb * NVT;     i += 256) a += nll_part[i];
    for (int i = tid; i < Bb;           i += 256) b += klt_part[i];
    for (int i = tid; i < KLA_BLOCKS;   i += 256) c += kla_part[i];
    red[tid] = a; __syncthreads();
    for (int o = 128; o > 0; o >>= 1) { if (tid < o) red[tid] += red[tid + o]; __syncthreads(); }
    const float snll = red[0]; __syncthreads();
    red[tid] = b; __syncthreads();
    for (int o = 128; o > 0; o >>= 1) { if (tid < o) red[tid] += red[tid + o]; __syncthreads(); }
    const float sklt = red[0]; __syncthreads();
    red[tid] = c; __syncthreads();
    for (int o = 128; o > 0; o >>= 1) { if (tid < o) red[tid] += red[tid + o]; __syncthreads(); }
    const float skla = red[0];
    if (tid == 0) {
        const float coeff = 100000.0f / (float)Bb;
        const float nll = snll * coeff;
        const float klt = sklt * coeff;
        const float ke  = kle[0];
        out[0] = nll + ke + klt + skla;
        out[1] = nll;
        out[2] = ke;
        out[3] = klt;
        out[4] = skla;
    }
}

// ---------------------------------------------------------------
extern "C" void kernel_launch(void* const* d_in, const int* in_sizes, int n_in,
                              void* d_out, int out_size, void* d_ws, size_t ws_size,
                              hipStream_t stream)
{
    (void)in_sizes; (void)n_in; (void)out_size; (void)ws_size;
    const float* bows     = (const float*)d_in[0];
    const int*   times    = (const int*)  d_in[1];
    const float* rnn_inp  = (const float*)d_in[2];
    const float* rho_W    = (const float*)d_in[3];
    const float* mu_q_a   = (const float*)d_in[4];
    const float* ls_q_a   = (const float*)d_in[5];
    const float* W1       = (const float*)d_in[6];
    const float* b1       = (const float*)d_in[7];
    const float* W2       = (const float*)d_in[8];
    const float* b2       = (const float*)d_in[9];
    const float* muthW    = (const float*)d_in[10];
    const float* muthb    = (const float*)d_in[11];
    const float* lsthW    = (const float*)d_in[12];
    const float* lsthb    = (const float*)d_in[13];
    const float* emapW    = (const float*)d_in[14];
    const float* emapb    = (const float*)d_in[15];
    const float* Wih      = (const float*)d_in[16];
    const float* Whh      = (const float*)d_in[17];
    const float* bih      = (const float*)d_in[18];
    const float* bhh      = (const float*)d_in[19];
    const float* muetaW   = (const float*)d_in[20];
    const float* muetab   = (const float*)d_in[21];
    const float* lsetaW   = (const float*)d_in[22];
    const float* lsetab   = (const float*)d_in[23];
    float* out = (float*)d_out;

    // workspace carve-up (256B aligned slices)
    char* wsb = (char*)d_ws;
    size_t off = 0;
    auto alloc = [&](size_t nfloats) -> float* {
        float* p = (float*)(wsb + off);
        off += ((nfloats * sizeof(float) + 255) / 256) * 256;
        return p;
    };
    float* buf0     = alloc((size_t)Tt * Hh);
    float* buf1     = alloc((size_t)Tt * Hh);
    float* etas     = alloc((size_t)Tt * Kk);
    float* kle      = alloc(4);
    float* Xcat     = alloc((size_t)Bb * (Vv + Kk));
    float* h1       = alloc((size_t)Bb * ENe);
    float* h2       = alloc((size_t)Bb * ENe);
    float* mu_t     = alloc((size_t)Bb * Kk);
    float* ls_t     = alloc((size_t)Bb * Kk);
    float* theta    = alloc((size_t)Bb * Kk);
    float* klt_part = alloc(Bb);
    float* kla_part = alloc(KLA_BLOCKS);
    float* nll_part = alloc((size_t)Bb * NVT);
    float* Achunk   = alloc((size_t)CTc * Kk * Rr);
    float* logitc   = alloc((size_t)CTc * Kk * Vv);
    float* rmax     = alloc(CTc * Kk);
    float* rinv     = alloc(CTc * Kk);

    const dim3 blk(32, 4);   // 4 waves/block, one 16x64 slab each
#define LAUNCH_GEMM(KD, A_, W_, BIAS_, C_, M_, N_, RELU_) \
    gemm_wmma<KD><<<dim3(((N_) + 255) / 256, ((M_) + 15) / 16), blk, 0, stream>>>( \
        (A_), (W_), (BIAS_), (C_), (M_), (N_), (RELU_))

    // 0) zero NLL partials (blocks for out-of-chunk b never write their slot)
    zero_kernel<<<(Bb * NVT + 255) / 256, 256, 0, stream>>>(nll_part, Bb * NVT);

    // 1) emb = rnn_inp @ eta_map_W^T + b   [T,H]
    LAUNCH_GEMM(Vv, rnn_inp, emapW, emapb, buf0, Tt, Hh, 0);

    // 2) 3-layer LSTM -> buf1 [T,H]
    lstm_kernel<<<1, 4 * Hh, 0, stream>>>(Wih, Whh, bih, bhh, buf0, buf1);

    // 3) eta scan -> etas [T,K], kl_eta
    eta_kernel<<<1, 128, 0, stream>>>(buf1, muetaW, muetab, lsetaW, lsetab, etas, kle);

    // 4) Xcat = [normalized bows, eta_td]
    xcat_kernel<<<Bb, 256, 0, stream>>>(bows, times, etas, Xcat);

    // 5) theta encoder (WMMA GEMMs)
    LAUNCH_GEMM(Vv + Kk, Xcat, W1, b1, h1, Bb, ENe, 1);
    LAUNCH_GEMM(ENe,     h1,   W2, b2, h2, Bb, ENe, 1);
    LAUNCH_GEMM(ENe,     h2, muthW, muthb, mu_t, Bb, Kk, 0);
    LAUNCH_GEMM(ENe,     h2, lsthW, lsthb, ls_t, Bb, Kk, 0);
    theta_kernel<<<Bb, 64, 0, stream>>>(mu_t, ls_t, times, etas, theta, klt_part);

    // 6) kl_alpha (fully vectorized)
    klalpha_kernel<<<KLA_BLOCKS, 256, 0, stream>>>(mu_q_a, ls_q_a, kla_part);

    // 7) beta logits + softmax + NLL in time chunks (15 GMAC WMMA GEMM total)
    for (int t0 = 0; t0 < Tt; t0 += CTc) {
        gather_alpha<<<(CTc * Kk * Rr + 255) / 256, 256, 0, stream>>>(mu_q_a, Achunk, t0);
        LAUNCH_GEMM(Rr, Achunk, rho_W, nullptr, logitc, CTc * Kk, Vv, 0);
        rowstats_kernel<<<CTc * Kk, 256, 0, stream>>>(logitc, rmax, rinv);
        nll_kernel<<<dim3(NVT, Bb), 256, 0, stream>>>(logitc, rmax, rinv, theta,
                                                      bows, times, nll_part, t0);
    }
#undef LAUNCH_GEMM

    // 8) deterministic final reduction -> 5 output scalars
    finalize_kernel<<<1, 256, 0, stream>>>(nll_part, klt_part, kla_part, kle, out);
}